// GCN_77180562309785
// MI455X (gfx1250) — compile-verified
//
#include <hip/hip_runtime.h>

#define NN 2048
#define EE 4
#define LL 3
#define EMB 128
#define NUM_TYPES 32
#define OUTW 32        // EMB/E
#define CHUNK 64       // K rows of x staged in LDS per iteration
#define PSTR 288       // LDS stride (floats) per K-pair row: 2*EMB + 32 (bank shift)
#define XSTR 140       // prop tile row stride (floats), reuses the x buffer
#define WSTR 132       // W tile row stride (floats) in LDS
#define XBUF_F ((CHUNK / 2) * PSTR)   // 9216 floats
#define WBUF_F (OUTW * WSTR)          // 4224 floats

typedef __attribute__((ext_vector_type(2))) float v2f;
typedef __attribute__((ext_vector_type(8))) float v8f;

#if defined(__has_builtin)
#if __has_builtin(__builtin_amdgcn_global_load_async_to_lds_b128)
#define HAVE_ASYNC_LDS 1
#endif
#endif

#ifdef HAVE_ASYNC_LDS
// Param 0 is 'int __vector_size__(16) addrspace(1)*' per clang's diagnostic;
// mirror for the LDS side in addrspace(3).
typedef int v4i __attribute__((vector_size(16)));
typedef __attribute__((address_space(1))) v4i v4i_g;
typedef __attribute__((address_space(3))) v4i v4i_l;
#endif

// -------------------------------------------------------------------------
// Kernel 1: per-row inverse L2 norm of A.  inv[e*N+n] = 1/max(||A[e,n,:]||,eps)
// One wave (32 lanes) per row.
// -------------------------------------------------------------------------
__global__ __launch_bounds__(256) void gcn_rownorm(const float* __restrict__ A,
                                                   float* __restrict__ inv) {
  int wave = (int)((blockIdx.x * blockDim.x + threadIdx.x) >> 5);
  int lane = (int)(threadIdx.x & 31);
  if (wave >= EE * NN) return;
  const float* row = A + (size_t)wave * NN;
  float s = 0.f;
  for (int i = lane; i < NN; i += 32) {
    float v = row[i];
    s += v * v;
  }
  #pragma unroll
  for (int off = 16; off > 0; off >>= 1) s += __shfl_xor(s, off, 32);
  if (lane == 0) {
    float nrm = sqrtf(s);
    inv[wave] = 1.0f / fmaxf(nrm, 1e-12f);
  }
}

// -------------------------------------------------------------------------
// Kernel 2: one GCN layer.
//   prop = diag(inv) * (A[e] @ xin)   (f32 WMMA 16x16x4; x staged in LDS with
//                                      K-pair interleave -> single ds_load_b64
//                                      per B operand, no packing movs)
//   h    = relu(prop @ W[e,l]^T)      (f32 WMMA 16x16x4; W async-staged in LDS)
//   xout[:, e*32:(e+1)*32] = h
// Block = 128 threads = 4 waves; each wave owns a 16-row strip of one edge.
// grid.x = E * (N/64) = 128.
// -------------------------------------------------------------------------
__global__ __launch_bounds__(128) void gcn_layer(const float* __restrict__ A,
                                                 const float* __restrict__ invA,
                                                 const float* __restrict__ xin,
                                                 const float* __restrict__ Wbase,
                                                 int l,
                                                 float* __restrict__ xout) {
  // [0, XBUF_F): x chunk, K-pair interleaved; reused later as prop tiles.
  // [XBUF_F, XBUF_F+WBUF_F): W[e,l] tile, async-staged, padded rows.
  __shared__ float lds[XBUF_F + WBUF_F];

  const int tid  = (int)threadIdx.x;
  const int wave = tid >> 5;
  const int lane = tid & 31;
  const int hl   = lane >> 4;   // 0: lanes 0-15, 1: lanes 16-31
  const int l16  = lane & 15;

  const int e       = (int)(blockIdx.x >> 5);  // edge 0..3
  const int rb      = (int)(blockIdx.x & 31);  // 64-row block
  const int rowbase = rb * 64 + wave * 16;

  const float* Ae   = A + (size_t)e * NN * NN;
  const float* arow = Ae + (size_t)(rowbase + l16) * NN;  // this lane's A row
  const float* Wle  = Wbase + ((size_t)e * LL + l) * OUTW * EMB;  // [32,128]
  float*       wlds = &lds[XBUF_F];

  // ---- async-stage W[e,l] into LDS; waited only after the K loop ----------
  #pragma unroll
  for (int j = 0; j < 8; ++j) {
    int idx = j * 128 + tid;            // float4 index 0..1023
    int r   = idx >> 5;                 // W row 0..31
    int c4  = idx & 31;                 // float4 column
#ifdef HAVE_ASYNC_LDS
    float* gm = (float*)(Wle + (size_t)r * EMB + c4 * 4);
    __builtin_amdgcn_global_load_async_to_lds_b128(
        (v4i_g*)gm, (v4i_l*)&wlds[r * WSTR + c4 * 4], 0, 0);
#else
    *(float4*)&wlds[r * WSTR + c4 * 4] =
        *(const float4*)(Wle + (size_t)r * EMB + c4 * 4);
#endif
  }

  v8f acc[8] = {};
  for (int kc = 0; kc < NN; kc += CHUNK) {
    // ---- stage x[kc:kc+64, :] K-pair interleaved:
    //      lds[(k/2)*PSTR + c*2 + (k&1)] = x[kc+k][c]
    #pragma unroll
    for (int j = 0; j < 16; ++j) {
      int idx = j * 128 + tid;          // 0..2047
      int p   = idx >> 6;               // K-pair row 0..31
      int c   = (idx & 63) * 2;         // even column 0..126
      const float* g0 = xin + (size_t)(kc + 2 * p) * EMB + c;
      float2 u = *(const float2*)g0;         // x[2p][c], x[2p][c+1]
      float2 v = *(const float2*)(g0 + EMB); // x[2p+1][c], x[2p+1][c+1]
      *(float4*)&lds[p * PSTR + c * 2] = make_float4(u.x, v.x, u.y, v.y);
    }
    __syncthreads();

    if (kc + CHUNK < NN) __builtin_prefetch(arow + kc + CHUNK, 0, 1);

    // ---- matmul 1: acc += A[rows, kc:kc+64] @ x_chunk --------------------
    for (int k0 = 0; k0 < CHUNK; k0 += 4) {
      // A operand 16x4 f32: VGPR0 = K(0|2), VGPR1 = K(1|3); M = lane%16.
      v2f a = *(const v2f*)(arow + kc + k0 + hl * 2);
      // B operand: {x[k][c], x[k+1][c]} is one contiguous b64 in LDS.
      const float* xk = &lds[((k0 >> 1) + hl) * PSTR + l16 * 2];
      #pragma unroll
      for (int ct = 0; ct < 8; ++ct) {
        v2f b = *(const v2f*)(xk + ct * 32);
        acc[ct] = __builtin_amdgcn_wmma_f32_16x16x4_f32(
            false, a, false, b, (short)0, acc[ct], false, false);
      }
    }
    __syncthreads();  // everyone done reading the chunk before re-fill/reuse
  }

  // ---- make W tile (and all LDS) visible; x buffer free for reuse ---------
#ifdef HAVE_ASYNC_LDS
#if defined(__has_builtin) && __has_builtin(__builtin_amdgcn_s_wait_asynccnt)
  __builtin_amdgcn_s_wait_asynccnt(0);
#else
  asm volatile("s_wait_asynccnt 0" ::: "memory");
#endif
#endif
  __syncthreads();

  // ---- row normalization (per-row scalar on D layout) ---------------------
  const float* inve = invA + e * NN;
  #pragma unroll
  for (int r = 0; r < 8; ++r) {
    float s = inve[rowbase + r + hl * 8];
    #pragma unroll
    for (int ct = 0; ct < 8; ++ct) acc[ct][r] *= s;
  }

  // ---- re-layout D-format -> A-operand format via LDS (reuse x buffer) ----
  float* P = &lds[wave * 16 * XSTR];
  #pragma unroll
  for (int ct = 0; ct < 8; ++ct) {
    #pragma unroll
    for (int r = 0; r < 8; ++r) {
      P[(r + hl * 8) * XSTR + ct * 16 + l16] = acc[ct][r];
    }
  }

  // ---- matmul 2: h[16,32] = prop @ W[e,l]^T (W from LDS) ------------------
  v8f h[2] = {};
  for (int d0 = 0; d0 < EMB; d0 += 4) {
    const int da = d0 + hl * 2;
    v2f a = *(const v2f*)(P + l16 * XSTR + da);  // prop rows as A operand
    #pragma unroll
    for (int ot = 0; ot < 2; ++ot) {
      v2f b = *(const v2f*)(&wlds[(ot * 16 + l16) * WSTR + da]);  // W^T tile
      h[ot] = __builtin_amdgcn_wmma_f32_16x16x4_f32(
          false, a, false, b, (short)0, h[ot], false, false);
    }
  }

  // ---- relu + scatter into xout[:, e*32 .. e*32+31] -----------------------
  #pragma unroll
  for (int ot = 0; ot < 2; ++ot) {
    #pragma unroll
    for (int r = 0; r < 8; ++r) {
      float v = fmaxf(h[ot][r], 0.f);
      xout[(size_t)(rowbase + r + hl * 8) * EMB + e * OUTW + ot * 16 + l16] = v;
    }
  }
}

// -------------------------------------------------------------------------
// Kernel 3: nodes_out = x @ final_W^T + final_b   ([2048,128]x[128,128])
// One wave per 16-row strip; 8 waves/block; grid = 16.
// -------------------------------------------------------------------------
__global__ __launch_bounds__(256) void gcn_final(const float* __restrict__ xin,
                                                 const float* __restrict__ FW,
                                                 const float* __restrict__ fb,
                                                 float* __restrict__ nodes_out) {
  const int wave = (int)(threadIdx.x >> 5);
  const int lane = (int)(threadIdx.x & 31);
  const int hl   = lane >> 4;
  const int l16  = lane & 15;
  const int rowbase = (int)blockIdx.x * 128 + wave * 16;

  v8f acc[8] = {};
  const float* xr = xin + (size_t)(rowbase + l16) * EMB;
  for (int d0 = 0; d0 < EMB; d0 += 4) {
    const int da = d0 + hl * 2;
    v2f a = *(const v2f*)(xr + da);
    #pragma unroll
    for (int ct = 0; ct < 8; ++ct) {
      v2f b = *(const v2f*)(FW + (size_t)(ct * 16 + l16) * EMB + da);
      acc[ct] = __builtin_amdgcn_wmma_f32_16x16x4_f32(
          false, a, false, b, (short)0, acc[ct], false, false);
    }
  }
  #pragma unroll
  for (int ct = 0; ct < 8; ++ct) {
    float bias = fb[ct * 16 + l16];
    #pragma unroll
    for (int r = 0; r < 8; ++r) {
      nodes_out[(size_t)(rowbase + r + hl * 8) * EMB + ct * 16 + l16] =
          acc[ct][r] + bias;
    }
  }
}

// -------------------------------------------------------------------------
// Kernel 4: build gse output [B=32, C=128, H=128, W=128].
// One wave per (b,h) row; each lane owns 4 consecutive w's -> b128 stores
// (512B contiguous per wave per channel).  Source rows are a ~24KB hot set.
// -------------------------------------------------------------------------
__global__ __launch_bounds__(128) void gcn_gse(const int* __restrict__ gs,
                                               const float* __restrict__ obj_emb,
                                               const float* __restrict__ nodes_out,
                                               float* __restrict__ out) {
  const int wave = (int)(threadIdx.x >> 5);
  const int lane = (int)(threadIdx.x & 31);
  const int bh   = (int)blockIdx.x * 4 + wave;  // b*128 + h
  const int b    = bh >> 7;
  const int h    = bh & 127;
  const int w0   = lane * 4;

  int4 t = *(const int4*)(gs + (size_t)bh * 128 + w0);
  const float* s0 = (t.x < (NUM_TYPES / 2)) ? nodes_out + (size_t)t.x * EMB
                                            : obj_emb + (size_t)t.x * EMB;
  const float* s1 = (t.y < (NUM_TYPES / 2)) ? nodes_out + (size_t)t.y * EMB
                                            : obj_emb + (size_t)t.y * EMB;
  const float* s2 = (t.z < (NUM_TYPES / 2)) ? nodes_out + (size_t)t.z * EMB
                                            : obj_emb + (size_t)t.z * EMB;
  const float* s3 = (t.w < (NUM_TYPES / 2)) ? nodes_out + (size_t)t.w * EMB
                                            : obj_emb + (size_t)t.w * EMB;

  float* o = out + (size_t)b * EMB * 128 * 128 + (size_t)h * 128 + w0;
  #pragma unroll 4
  for (int c = 0; c < EMB; ++c) {
    float4 v = make_float4(s0[c], s1[c], s2[c], s3[c]);
    *(float4*)(o + (size_t)c * 128 * 128) = v;
  }
}

// -------------------------------------------------------------------------
extern "C" void kernel_launch(void* const* d_in, const int* in_sizes, int n_in,
                              void* d_out, int out_size, void* d_ws, size_t ws_size,
                              hipStream_t stream) {
  const int*   game_state = (const int*)d_in[0];    // [32,128,128]
  const float* A          = (const float*)d_in[1];  // [4,2048,2048]
  const float* node_emb   = (const float*)d_in[2];  // [2048,128]
  const float* obj_emb    = (const float*)d_in[3];  // [32,128]
  const float* W          = (const float*)d_in[4];  // [4,3,32,128]
  const float* final_W    = (const float*)d_in[5];  // [128,128]
  const float* final_b    = (const float*)d_in[6];  // [128]

  float* out_gse   = (float*)d_out;                           // 32*128*128*128
  float* nodes_out = out_gse + (size_t)32 * 128 * 128 * 128;  // 2048*128

  float* invA = (float*)d_ws;                 // E*N floats
  float* x1   = invA + EE * NN;               // N*EMB
  float* x2   = x1 + (size_t)NN * EMB;        // N*EMB

  gcn_rownorm<<<EE * NN / 8, 256, 0, stream>>>(A, invA);
  gcn_layer<<<128, 128, 0, stream>>>(A, invA, node_emb, W, 0, x1);
  gcn_layer<<<128, 128, 0, stream>>>(A, invA, x1, W, 1, x2);
  gcn_layer<<<128, 128, 0, stream>>>(A, invA, x2, W, 2, x1);
  gcn_final<<<16, 256, 0, stream>>>(x1, final_W, final_b, nodes_out);
  gcn_gse<<<32 * 128 / 4, 128, 0, stream>>>(game_state, obj_emb, nodes_out, out_gse);
}